// HopAttentionLayer_16063177687233
// MI455X (gfx1250) — compile-verified
//
#include <hip/hip_runtime.h>

// Problem constants (match reference)
constexpr int kB = 256;
constexpr int kT = 64;
constexpr int kN = 2048;
constexpr int kD = 128;

constexpr int kRowStride = kD + 4;  // 132 floats: 16B-aligned rows, bank-skewed by 4

typedef __attribute__((ext_vector_type(2))) float v2f;
typedef __attribute__((ext_vector_type(8))) float v8f;

// ---------------------------------------------------------------------------
// Kernel 1: per-batch scores via f32 WMMA GEMV + softmax -> attention (d_ws)
// One block per batch, 128 threads = 4 waves (wave32). Each wave owns 512 rows.
// ---------------------------------------------------------------------------
__global__ __launch_bounds__(128) void hop_attn_scores_kernel(
    const float* __restrict__ tgt,   // [B,T,D]
    const float* __restrict__ ctx,   // [B,N,D]
    const float* __restrict__ W,     // [2D]
    const float* __restrict__ bias,  // [1]
    float* __restrict__ att)         // [B,N] normalized attention out
{
  __shared__ float s_w[2 * kD];                    // 1 KB
  __shared__ float s_scores[kN];                   // 8 KB
  __shared__ float s_tile[4][16 * kRowStride];     // 33 KB (per-wave 16x128 tiles)
  __shared__ float s_red[128];
  __shared__ float s_misc[1];

  const int tid  = threadIdx.x;
  const int b    = blockIdx.x;
  const int lane = tid & 31;
  const int wave = tid >> 5;
  const int half = lane >> 4;   // 0: K pair {0,1}; 1: K pair {2,3} within a 4-chunk
  const int m    = lane & 15;   // A-matrix row within tile

  // Stage W into LDS (Wt = s_w[0..127], Wc = s_w[128..255])
  s_w[tid]       = W[tid];
  s_w[tid + 128] = W[tid + 128];
  __syncthreads();

  // ---- Phase A: tgt_term = sum_{t,d} tgt[b,t,d] * Wt[d]  (d == tid since bdim==128)
  {
    const float wt = s_w[tid];
    const float* p = tgt + (size_t)b * kT * kD;
    float acc = 0.f;
#pragma unroll 4
    for (int i = tid; i < kT * kD; i += 128) acc += p[i] * wt;
    s_red[tid] = acc;
    __syncthreads();
    for (int s = 64; s > 0; s >>= 1) {
      if (tid < s) s_red[tid] += s_red[tid + s];
      __syncthreads();
    }
    if (tid == 0) s_misc[0] = s_red[0];
    __syncthreads();
  }
  const float tgt_term = s_misc[0];
  const float b0 = bias[0];
  __syncthreads();

  // ---- Phase B: ctx dots via V_WMMA_F32_16X16X4_F32, 16 rows per tile
  {
    float* tile = s_tile[wave];
    const float* wc = s_w + kD;

    for (int t = 0; t < 32; ++t) {
      const int n0 = wave * 512 + t * 16;
      const float* src = ctx + ((size_t)b * kN + n0) * kD;

      // Coalesced load: one 512B row per iteration (32 lanes x float4)
#pragma unroll
      for (int r = 0; r < 16; ++r) {
        const float4 v = *(const float4*)(src + r * kD + lane * 4);
        *(float4*)(tile + r * kRowStride + lane * 4) = v;
      }
      asm volatile("s_wait_dscnt 0" ::: "memory");
      __builtin_amdgcn_wave_barrier();

      // A: 16x4 f32 tiles from LDS (lane m = row; half selects K sub-pair)
      // B: Wc replicated across all 16 columns -> every D column = row dot
      v8f acc0 = {};
      v8f acc1 = {};
#pragma unroll
      for (int c = 0; c < 32; c += 2) {
        {
          const int k = c * 4 + half * 2;
          v2f a;  a.x = tile[m * kRowStride + k];  a.y = tile[m * kRowStride + k + 1];
          v2f bb; bb.x = wc[k];                    bb.y = wc[k + 1];
          acc0 = __builtin_amdgcn_wmma_f32_16x16x4_f32(
              false, a, false, bb, (short)0, acc0, false, false);
        }
        {
          const int k = (c + 1) * 4 + half * 2;
          v2f a;  a.x = tile[m * kRowStride + k];  a.y = tile[m * kRowStride + k + 1];
          v2f bb; bb.x = wc[k];                    bb.y = wc[k + 1];
          acc1 = __builtin_amdgcn_wmma_f32_16x16x4_f32(
              false, a, false, bb, (short)0, acc1, false, false);
        }
      }

      // D layout: VGPR r, lanes 0-15 -> M=r, lanes 16-31 -> M=r+8 (all N equal).
      // Lanes 0 and 16 write the 16 scores.
      if (m == 0) {
#pragma unroll
        for (int r = 0; r < 8; ++r) {
          const float dot = acc0[r] + acc1[r];
          s_scores[n0 + half * 8 + r] = tgt_term + (float)kT * (dot + b0);
        }
      }
      __builtin_amdgcn_wave_barrier();
    }
  }
  __syncthreads();

  // ---- Phase C: softmax over the 2048 scores
  float lmax = -3.402823466e38f;
  for (int i = tid; i < kN; i += 128) lmax = fmaxf(lmax, s_scores[i]);
  s_red[tid] = lmax;
  __syncthreads();
  for (int s = 64; s > 0; s >>= 1) {
    if (tid < s) s_red[tid] = fmaxf(s_red[tid], s_red[tid + s]);
    __syncthreads();
  }
  const float gmax = s_red[0];
  __syncthreads();

  float lsum = 0.f;
  for (int i = tid; i < kN; i += 128) {
    const float e = __expf(s_scores[i] - gmax);
    s_scores[i] = e;
    lsum += e;
  }
  s_red[tid] = lsum;
  __syncthreads();
  for (int s = 64; s > 0; s >>= 1) {
    if (tid < s) s_red[tid] += s_red[tid + s];
    __syncthreads();
  }
  const float inv = 1.0f / s_red[0];

  for (int i = tid; i < kN; i += 128)
    att[(size_t)b * kN + i] = s_scores[i] * inv;
}

// ---------------------------------------------------------------------------
// Kernel 2: out[b,n,:] = att[b,n] * ctx[b,n,:]  (dominant 512MB stream)
// ---------------------------------------------------------------------------
__global__ __launch_bounds__(256) void hop_attn_scale_kernel(
    const float* __restrict__ ctx, const float* __restrict__ att,
    float* __restrict__ out)
{
  const size_t total4 = (size_t)kB * kN * kD / 4;  // 16,777,216 float4
  size_t i = (size_t)blockIdx.x * blockDim.x + threadIdx.x;
  const size_t stride = (size_t)gridDim.x * blockDim.x;
  for (; i < total4; i += stride) {
    const float a = att[i >> 5];  // (i*4)/128 == flattened (b*N + n)
    float4 v = ((const float4*)ctx)[i];
    v.x *= a; v.y *= a; v.z *= a; v.w *= a;
    ((float4*)out)[i] = v;
  }
}

extern "C" void kernel_launch(void* const* d_in, const int* in_sizes, int n_in,
                              void* d_out, int out_size, void* d_ws, size_t ws_size,
                              hipStream_t stream) {
  const float* tgt  = (const float*)d_in[0];  // [B,T,D]
  const float* ctx  = (const float*)d_in[1];  // [B,N,D]
  const float* W    = (const float*)d_in[2];  // [2D]
  const float* bias = (const float*)d_in[3];  // [1]
  float* att = (float*)d_ws;                  // [B,N] scratch (2 MB)
  float* out = (float*)d_out;                 // [B,N,D]

  hop_attn_scores_kernel<<<dim3(kB), dim3(128), 0, stream>>>(tgt, ctx, W, bias, att);
  hop_attn_scale_kernel<<<dim3(4096), dim3(256), 0, stream>>>(ctx, att, out);
}